// GCNLayer_24644522345228
// MI455X (gfx1250) — compile-verified
//
#include <hip/hip_runtime.h>

typedef __attribute__((ext_vector_type(2))) float v2f;
typedef __attribute__((ext_vector_type(8))) float v8f;

#define IN_DIM 64
#define OUT_DIM 32
#define LDS_STRIDE 68   // 64 + 4-float pad: conflict-free ds_load_b64 across a wave

// ---------------------------------------------------------------------------
// Kernel 1: h = x @ W + b using V_WMMA_F32_16X16X4_F32.
// One wave32 computes a 16x32 output tile (two 16x16 accumulators).
// Block = 256 threads = 8 waves = 128 rows.
// W is staged TRANSPOSED in LDS so each B fragment is one ds_load_b64.
// ---------------------------------------------------------------------------
__global__ __launch_bounds__(256) void gcn_gemm_wmma(
    const float* __restrict__ x, const float* __restrict__ W,
    const float* __restrict__ bias, float* __restrict__ h, int n_nodes)
{
    __shared__ float sWt[OUT_DIM * LDS_STRIDE];  // W^T with padded stride (~8.5 KB)
    __shared__ float sB[OUT_DIM];

    for (int i = threadIdx.x; i < IN_DIM * OUT_DIM; i += 256) {
        const int k = i >> 5;       // W is [64][32] row-major: i = k*32 + n
        const int n = i & 31;
        sWt[n * LDS_STRIDE + k] = W[i];
    }
    if (threadIdx.x < OUT_DIM) sB[threadIdx.x] = bias[threadIdx.x];
    __syncthreads();

    const int wave = threadIdx.x >> 5;
    const int lane = threadIdx.x & 31;
    const int row0 = (blockIdx.x * 8 + wave) * 16;
    if (row0 >= n_nodes) return;          // wave-uniform: EXEC stays all-1s

    // A-matrix 16x4 f32 layout (ISA 7.12.2):
    //   lanes 0-15 : row M=lane,    VGPR0=K0, VGPR1=K1
    //   lanes 16-31: row M=lane-16, VGPR0=K2, VGPR1=K3
    const int lhalf = lane >> 4;          // 0 or 1
    const int lsub  = lane & 15;          // row (A) / column (B,C,D)
    const float* xrow  = x + (size_t)(row0 + lsub) * IN_DIM + 2 * lhalf;
    const float* bptr0 = sWt + lsub * LDS_STRIDE + 2 * lhalf;         // N = lsub
    const float* bptr1 = sWt + (16 + lsub) * LDS_STRIDE + 2 * lhalf;  // N = 16+lsub

    v8f acc0 = {};   // N = 0..15
    v8f acc1 = {};   // N = 16..31

#pragma unroll
    for (int k = 0; k < IN_DIM; k += 4) {
        const v2f a  = *(const v2f*)(xrow + k);    // global_load_b64
        const v2f b0 = *(const v2f*)(bptr0 + k);   // ds_load_b64 (contiguous K pair)
        const v2f b1 = *(const v2f*)(bptr1 + k);   // ds_load_b64
        acc0 = __builtin_amdgcn_wmma_f32_16x16x4_f32(
            false, a, false, b0, (short)0, acc0, false, false);
        acc1 = __builtin_amdgcn_wmma_f32_16x16x4_f32(
            false, a, false, b1, (short)0, acc1, false, false);
    }

    // C/D layout: VGPR i -> (M = 8*lhalf + i, N = lsub)
    const float bb0 = sB[lsub];
    const float bb1 = sB[16 + lsub];
#pragma unroll
    for (int i = 0; i < 8; ++i) {
        const int m = 8 * lhalf + i;
        float* o = h + (size_t)(row0 + m) * OUT_DIM + lsub;
        o[0]  = acc0[i] + bb0;
        o[16] = acc1[i] + bb1;
    }
}

// ---------------------------------------------------------------------------
// Kernel 2: zero the output (harness poisons d_out with 0xAA).
// ---------------------------------------------------------------------------
__global__ __launch_bounds__(256) void gcn_zero(float* __restrict__ out, int n)
{
    const int i = blockIdx.x * 256 + threadIdx.x;
    if (i < n) out[i] = 0.0f;
}

// ---------------------------------------------------------------------------
// Kernel 3: out[dst] += h[src] * w  (COO scatter-add).
// Each thread: 1 edge x 4 features -> 128-bit gather + 4 f32 atomics.
// h and out are L2-resident (12.8 MB each vs 192 MB L2).
// ---------------------------------------------------------------------------
__global__ __launch_bounds__(256) void gcn_scatter(
    const float* __restrict__ h, const int* __restrict__ esrc,
    const int* __restrict__ edst, const float* __restrict__ ew,
    float* __restrict__ out, int n_edges)
{
    const long long tid = (long long)blockIdx.x * 256 + threadIdx.x;
    const int e = (int)(tid >> 3);
    const int f = ((int)tid & 7) * 4;
    if (e >= n_edges) return;

    const int   s = esrc[e];
    const int   d = edst[e];
    const float w = ew[e];

    const float4 v = *(const float4*)(h + (size_t)s * OUT_DIM + f);
    float* o = out + (size_t)d * OUT_DIM + f;
    atomicAdd(o + 0, v.x * w);
    atomicAdd(o + 1, v.y * w);
    atomicAdd(o + 2, v.z * w);
    atomicAdd(o + 3, v.w * w);
}

// ---------------------------------------------------------------------------
extern "C" void kernel_launch(void* const* d_in, const int* in_sizes, int n_in,
                              void* d_out, int out_size, void* d_ws, size_t ws_size,
                              hipStream_t stream)
{
    const float* x    = (const float*)d_in[0];
    const int*   esrc = (const int*)d_in[1];
    const int*   edst = (const int*)d_in[2];
    const float* ew   = (const float*)d_in[3];
    const float* W    = (const float*)d_in[4];
    const float* bias = (const float*)d_in[5];
    float* out = (float*)d_out;
    float* h   = (float*)d_ws;   // n_nodes * 32 floats = 12.8 MB scratch

    const int n_nodes = in_sizes[0] / IN_DIM;
    const int n_edges = in_sizes[1];

    // 1) GEMM: 128 rows per block
    dim3 g1((unsigned)((n_nodes + 127) / 128));
    gcn_gemm_wmma<<<g1, 256, 0, stream>>>(x, W, bias, h, n_nodes);

    // 2) zero output
    dim3 g2((unsigned)((out_size + 255) / 256));
    gcn_zero<<<g2, 256, 0, stream>>>(out, out_size);

    // 3) scatter-add: 8 threads per edge
    const long long sthreads = (long long)n_edges * 8;
    dim3 g3((unsigned)((sthreads + 255) / 256));
    gcn_scatter<<<g3, 256, 0, stream>>>(h, esrc, edst, ew, out, n_edges);
}